// HolographicLayer_41455024341724
// MI455X (gfx1250) — compile-verified
//
#include <hip/hip_runtime.h>

// eta = r2v . ccorr_half(s2v, o2v)  with d = 201, pad = 100
// Rewritten as eta = r^T * S * o, S[k,i] = s_pad[k + i]  (s_pad zero-padded, offset 100)
// Tiled 13x13 over 16x16 tiles, computed with V_WMMA_F32_16X16X4_F32.

typedef __attribute__((ext_vector_type(2))) float v2f;
typedef __attribute__((ext_vector_type(8))) float v8f;

#define DIMD 201
#define PADC 100
#define DT   208   // 13 * 16
#define SLEN 416   // s_lds[j] = s2v[j-100], j in [0,416); max index used: 414

__global__ __launch_bounds__(256) void holo_wmma_kernel(const int* __restrict__ x,
                                                        const float* __restrict__ E,
                                                        const float* __restrict__ R,
                                                        float* __restrict__ out) {
    __shared__ float s_lds[SLEN];
    __shared__ float o_lds[DT];
    __shared__ float r_lds[DT];
    __shared__ float eta_acc;

    const int tid = threadIdx.x;

    // Only row 0 of x is used by the reference.
    const int si = x[0];
    const int oi = x[1];
    const int pi = x[2];

    const float* __restrict__ srow = E + (long long)si * DIMD;
    const float* __restrict__ orow = E + (long long)oi * DIMD;
    const float* __restrict__ rrow = R + (long long)pi * DIMD;

    // Stage padded vectors into LDS. Zero padding makes all OOB terms exact 0.
    for (int j = tid; j < SLEN; j += 256) {
        const int g = j - PADC;
        s_lds[j] = (g >= 0 && g < DIMD) ? srow[g] : 0.0f;
    }
    for (int j = tid; j < DT; j += 256) {
        o_lds[j] = (j < DIMD) ? orow[j] : 0.0f;
        r_lds[j] = (j < DIMD) ? rrow[j] : 0.0f;
    }
    if (tid == 0) eta_acc = 0.0f;
    __syncthreads();

    const int lane = tid & 31;
    const int wave = tid >> 5;
    const int m    = lane & 15;              // M row (A) / N col (C) within half-wave
    const int kofs = (lane >= 16) ? 2 : 0;   // f32 A/B layout: lanes 16-31 carry K=2,3

    // 13 k-tiles spread over 8 waves; loop bound is wave-uniform so EXEC is
    // all-ones around every WMMA (ISA requirement).
    for (int kt = wave; kt < 13; kt += 8) {
        const int kbase = kt * 16;
        v8f c = {0.f, 0.f, 0.f, 0.f, 0.f, 0.f, 0.f, 0.f};

        for (int it = 0; it < 13; ++it) {
            #pragma unroll
            for (int st = 0; st < 4; ++st) {
                const int ib = it * 16 + st * 4 + kofs; // this lane's K base
                v2f a, b;
                // A: 16x4 Toeplitz slab  A[m][K] = s_pad[kbase + m + i]
                a.x = s_lds[kbase + m + ib];
                a.y = s_lds[kbase + m + ib + 1];
                // B: 4x16 with o broadcast across all N columns (value depends
                // only on the K row for this lane group).
                b.x = o_lds[ib];
                b.y = o_lds[ib + 1];
                // (neg_a, A, neg_b, B, c_mod, C, reuse_a, reuse_b)
                c = __builtin_amdgcn_wmma_f32_16x16x4_f32(
                        false, a, false, b, (short)0, c, false, false);
            }
        }

        // C layout: VGPR v holds (M = v + 8*(lane>=16), N = lane&15).
        // Column N=0 (the mat-vec result y) lives in lanes 0 and 16.
        if (m == 0) {
            const int mo = (lane >= 16) ? 8 : 0;
            float p = 0.0f;
            p += r_lds[kbase + mo + 0] * c[0];
            p += r_lds[kbase + mo + 1] * c[1];
            p += r_lds[kbase + mo + 2] * c[2];
            p += r_lds[kbase + mo + 3] * c[3];
            p += r_lds[kbase + mo + 4] * c[4];
            p += r_lds[kbase + mo + 5] * c[5];
            p += r_lds[kbase + mo + 6] * c[6];
            p += r_lds[kbase + mo + 7] * c[7];
            atomicAdd(&eta_acc, p);   // ds_add_f32
        }
    }

    __syncthreads();
    if (tid == 0) out[0] = eta_acc;
}

extern "C" void kernel_launch(void* const* d_in, const int* in_sizes, int n_in,
                              void* d_out, int out_size, void* d_ws, size_t ws_size,
                              hipStream_t stream) {
    (void)in_sizes; (void)n_in; (void)out_size; (void)d_ws; (void)ws_size;
    const int*   x = (const int*)d_in[0];
    const float* E = (const float*)d_in[1];
    const float* R = (const float*)d_in[2];
    float* out = (float*)d_out;

    // Scalar output, ~2.4 KB of input: latency-bound -> single workgroup,
    // 8 wave32 waves on one WGP, everything staged in LDS.
    holo_wmma_kernel<<<1, 256, 0, stream>>>(x, E, R, out);
}